// MedicalHGT_13056700580221
// MI455X (gfx1250) — compile-verified
//
#include <hip/hip_runtime.h>
#include <hip/hip_bf16.h>

typedef float v2f __attribute__((ext_vector_type(2)));
typedef float v8f __attribute__((ext_vector_type(8)));

// ---------------------------------------------------------------------------
// CDNA5 fp32 WMMA: D(16x16) = A(16x4) * B(4x16) + C, full fp32 precision.
// A layout: lane l -> row M=(l&15), K = 2*(l>>4) + {0,1} in {a.x, a.y}
// B layout: lane l -> col N=(l&15), K = 2*(l>>4) + {0,1} in {b.x, b.y}
// C/D layout: lane l, vgpr r -> M = r + 8*(l>>4), N = (l&15)
// ---------------------------------------------------------------------------
__device__ inline v8f wmma4(v2f a, v2f b, v8f c) {
  return __builtin_amdgcn_wmma_f32_16x16x4_f32(
      /*neg_a=*/false, a, /*neg_b=*/false, b,
      /*c_mod=*/(short)0, c, /*reuse_a=*/false, /*reuse_b=*/false);
}

// monotonic uint encoding of float for atomicMax-based segment max
__device__ inline unsigned encf(float f) {
  unsigned u = __float_as_uint(f);
  return (u & 0x80000000u) ? ~u : (u | 0x80000000u);
}
__device__ inline float decf(unsigned u) {
  unsigned v = (u & 0x80000000u) ? (u & 0x7FFFFFFFu) : ~u;
  return __uint_as_float(v);
}

// ---------------------------------------------------------------------------
__global__ void k_fill_u32(unsigned* __restrict__ p, unsigned v, size_t n) {
  size_t i = (size_t)blockIdx.x * blockDim.x + threadIdx.x;
  if (i < n) p[i] = v;
}

// GEMM1: Y[N,64] = relu(X[N,128] @ W[128,64] + b). grid.x=N/16, block=128 (4 waves)
__global__ void k_in_proj(const float* __restrict__ X, const float* __restrict__ W,
                          const float* __restrict__ b, float* __restrict__ Y, int N) {
  int wave = threadIdx.x >> 5;
  int lane = threadIdx.x & 31;
  int row0 = blockIdx.x * 16;
  int col0 = wave * 16;
  int m  = lane & 15;
  int kb = (lane >> 4) << 1;
  v8f acc = {};
  const float* xrow = X + (size_t)(row0 + m) * 128;
  for (int k0 = 0; k0 < 128; k0 += 4) {
    v2f a; a.x = xrow[k0 + kb]; a.y = xrow[k0 + kb + 1];
    v2f bv; bv.x = W[(size_t)(k0 + kb) * 64 + col0 + m];
            bv.y = W[(size_t)(k0 + kb + 1) * 64 + col0 + m];
    acc = wmma4(a, bv, acc);
  }
  int col = col0 + m;
  int rbase = row0 + ((lane >> 4) << 3);
  float bias = b[col];
  for (int r = 0; r < 8; ++r) {
    float v = acc[r] + bias;
    Y[(size_t)(rbase + r) * 64 + col] = fmaxf(v, 0.f);
  }
}

// BN stats partial reduce. grid=256, block=256 (4 row-groups x 64 channels)
__global__ void k_bn_reduce(const float* __restrict__ Y, int N,
                            float* __restrict__ sum, float* __restrict__ sumsq) {
  int c = threadIdx.x & 63;
  int rg = threadIdx.x >> 6;
  float s = 0.f, ss = 0.f;
  for (int row = blockIdx.x * 4 + rg; row < N; row += gridDim.x * 4) {
    float v = Y[(size_t)row * 64 + c];
    s += v; ss += v * v;
  }
  __shared__ float ls[256], lss[256];
  ls[threadIdx.x] = s; lss[threadIdx.x] = ss;
  __syncthreads();
  if (rg == 0) {
    s  = ls[c]  + ls[64 + c]  + ls[128 + c]  + ls[192 + c];
    ss = lss[c] + lss[64 + c] + lss[128 + c] + lss[192 + c];
    atomicAdd(&sum[c], s);
    atomicAdd(&sumsq[c], ss);
  }
}

__global__ void k_bn_finalize(const float* __restrict__ sum, const float* __restrict__ sumsq,
                              float* __restrict__ mu, float* __restrict__ rs,
                              int N0, int N1, int N2) {
  int i = threadIdx.x;
  if (i >= 192) return;
  int t = i >> 6;
  float n = (float)(t == 0 ? N0 : (t == 1 ? N1 : N2));
  float m = sum[i] / n;
  float var = sumsq[i] / n - m * m;
  mu[i] = m;
  rs[i] = rsqrtf(var + 1e-5f);
}

__global__ void k_bn_apply(float* __restrict__ Y, long long n,
                           const float* __restrict__ gamma, const float* __restrict__ beta,
                           const float* __restrict__ mu, const float* __restrict__ rs) {
  long long i = (long long)blockIdx.x * blockDim.x + threadIdx.x;
  if (i >= n) return;
  int c = (int)(i & 63);
  Y[i] = gamma[c] * (Y[i] - mu[c]) * rs[c] + beta[c];
}

// GEMM2: KQV[N,192] = H[N,64] @ W[64,192] + b. grid=(N/16,3), block=128
__global__ void k_kqv(const float* __restrict__ Hm, const float* __restrict__ W,
                      const float* __restrict__ b, float* __restrict__ KQV, int N) {
  int wave = threadIdx.x >> 5;
  int lane = threadIdx.x & 31;
  int row0 = blockIdx.x * 16;
  int col0 = (blockIdx.y * 4 + wave) * 16;
  int m  = lane & 15;
  int kb = (lane >> 4) << 1;
  v8f acc = {};
  const float* hrow = Hm + (size_t)(row0 + m) * 64;
  for (int k0 = 0; k0 < 64; k0 += 4) {
    v2f a; a.x = hrow[k0 + kb]; a.y = hrow[k0 + kb + 1];
    v2f bv; bv.x = W[(size_t)(k0 + kb) * 192 + col0 + m];
            bv.y = W[(size_t)(k0 + kb + 1) * 192 + col0 + m];
    acc = wmma4(a, bv, acc);
  }
  int col = col0 + m;
  int rbase = row0 + ((lane >> 4) << 3);
  float bias = b[col];
  for (int r = 0; r < 8; ++r)
    KQV[(size_t)(rbase + r) * 192 + col] = acc[r] + bias;
}

// Edge pass 1: scores + per-(dst,head) running max. block=256 = 4 edges x 2 heads x 32 lanes
__global__ void k_edge_pass1(const int* __restrict__ src, const int* __restrict__ dst, int E,
                             const float* __restrict__ kqv_s, const float* __restrict__ kqv_d,
                             const float* __restrict__ Ak, const float* __restrict__ prel,
                             float* __restrict__ sc_out, unsigned* __restrict__ m_arr) {
  __shared__ float sA[2048];
  __shared__ float sK[4][64];
  for (int i = threadIdx.x; i < 2048; i += 256) sA[i] = Ak[i];
  int eg = threadIdx.x >> 6;
  int e = blockIdx.x * 4 + eg;
  int h = (threadIdx.x >> 5) & 1;
  int f = threadIdx.x & 31;
  bool valid = (e < E);
  int s = valid ? src[e] : 0;
  int d = valid ? dst[e] : 0;
  sK[eg][h * 32 + f] = kqv_s[(size_t)s * 192 + h * 32 + f];
  __syncthreads();
  float kr = 0.f;
  const float* kv = &sK[eg][h * 32];
  const float* Ah = &sA[h * 1024 + f];
#pragma unroll
  for (int dd = 0; dd < 32; ++dd) kr = fmaf(kv[dd], Ah[dd * 32], kr);
  float q = kqv_d[(size_t)d * 192 + 64 + h * 32 + f];
  float prod = q * kr;
#pragma unroll
  for (int o = 16; o; o >>= 1) prod += __shfl_xor(prod, o, 32);
  if (f == 0 && valid) {
    float sc = prod * prel[h] * 0.17677669529663687f;  // 1/sqrt(32)
    sc_out[(size_t)e * 2 + h] = sc;
    atomicMax(&m_arr[(size_t)d * 2 + h], encf(sc));
  }
}

// Edge pass 2: w = exp(sc-m); accumulate sum and weighted values.
__global__ void k_edge_pass2(const int* __restrict__ src, const int* __restrict__ dst, int E,
                             const float* __restrict__ kqv_s, const float* __restrict__ Av,
                             const float* __restrict__ sc_in, const unsigned* __restrict__ m_arr,
                             float* __restrict__ s_arr, float* __restrict__ agg) {
  __shared__ float sA[2048];
  __shared__ float sV[4][64];
  for (int i = threadIdx.x; i < 2048; i += 256) sA[i] = Av[i];
  int eg = threadIdx.x >> 6;
  int e = blockIdx.x * 4 + eg;
  int h = (threadIdx.x >> 5) & 1;
  int f = threadIdx.x & 31;
  bool valid = (e < E);
  int s = valid ? src[e] : 0;
  int d = valid ? dst[e] : 0;
  sV[eg][h * 32 + f] = kqv_s[(size_t)s * 192 + 128 + h * 32 + f];
  __syncthreads();
  float vr = 0.f;
  const float* vv = &sV[eg][h * 32];
  const float* Ah = &sA[h * 1024 + f];
#pragma unroll
  for (int dd = 0; dd < 32; ++dd) vr = fmaf(vv[dd], Ah[dd * 32], vr);
  if (valid) {
    float sc = sc_in[(size_t)e * 2 + h];
    float mval = decf(m_arr[(size_t)d * 2 + h]);
    float w = expf(sc - mval);
    if (f == 0) atomicAdd(&s_arr[(size_t)d * 2 + h], w);
    atomicAdd(&agg[(size_t)d * 64 + h * 32 + f], w * vr);
  }
}

__device__ inline float gelu_in(const float* __restrict__ agg, const float* __restrict__ sarr,
                                long long row, int k) {
  float den = sarr[row * 2 + (k >> 5)];
  float x = (den > 0.f) ? agg[row * 64 + k] / den : 0.f;
  return 0.5f * x * (1.f + erff(x * 0.70710678118654752f));
}

// GEMM3 + skip blend. grid=N/16, block=128
__global__ void k_out(const float* __restrict__ agg, const float* __restrict__ sarr,
                      const float* __restrict__ Hm, const float* __restrict__ W,
                      const float* __restrict__ b, const float* __restrict__ skipg,
                      float* __restrict__ Z, int N) {
  int wave = threadIdx.x >> 5;
  int lane = threadIdx.x & 31;
  int row0 = blockIdx.x * 16;
  int col0 = wave * 16;
  int m  = lane & 15;
  int kb = (lane >> 4) << 1;
  long long row = row0 + m;
  v8f acc = {};
  for (int k0 = 0; k0 < 64; k0 += 4) {
    v2f a; a.x = gelu_in(agg, sarr, row, k0 + kb);
           a.y = gelu_in(agg, sarr, row, k0 + kb + 1);
    v2f bv; bv.x = W[(size_t)(k0 + kb) * 64 + col0 + m];
            bv.y = W[(size_t)(k0 + kb + 1) * 64 + col0 + m];
    acc = wmma4(a, bv, acc);
  }
  float g = 1.f / (1.f + expf(-skipg[0]));
  int col = col0 + m;
  long long rbase = row0 + ((lane >> 4) << 3);
  float bias = b[col];
  for (int r = 0; r < 8; ++r) {
    long long rr = rbase + r;
    float aval = acc[r] + bias;
    Z[rr * 64 + col] = g * aval + (1.f - g) * Hm[rr * 64 + col];
  }
}

__global__ void k_pred(const int* __restrict__ qi, const int* __restrict__ ai, int n,
                       const float* __restrict__ zq, const float* __restrict__ za,
                       float* __restrict__ out) {
  int i = blockIdx.x * blockDim.x + threadIdx.x;
  if (i >= n) return;
  const float* a = zq + (long long)qi[i] * 64;
  const float* b = za + (long long)ai[i] * 64;
  float s = 0.f;
#pragma unroll
  for (int j = 0; j < 64; ++j) s = fmaf(a[j], b[j], s);
  out[i] = s;
}

// ---------------------------------------------------------------------------
extern "C" void kernel_launch(void* const* d_in, const int* in_sizes, int n_in,
                              void* d_out, int out_size, void* d_ws, size_t ws_size,
                              hipStream_t stream) {
  (void)in_sizes; (void)n_in; (void)out_size; (void)ws_size;

  const float* xq = (const float*)d_in[0];
  const float* xa = (const float*)d_in[1];
  const float* xe = (const float*)d_in[2];
  const int* eptr[4] = {(const int*)d_in[3], (const int*)d_in[4],
                        (const int*)d_in[5], (const int*)d_in[6]};
  const int* pos_idx = (const int*)d_in[7];
  const int* neg_idx = (const int*)d_in[8];
  const float* W_in     = (const float*)d_in[9];
  const float* b_in     = (const float*)d_in[10];
  const float* bn_gamma = (const float*)d_in[11];
  const float* bn_beta  = (const float*)d_in[12];
  const float* W_kqv    = (const float*)d_in[13];
  const float* b_kqv    = (const float*)d_in[14];
  const float* A_k      = (const float*)d_in[15];
  const float* A_v      = (const float*)d_in[16];
  const float* p_rel    = (const float*)d_in[17];
  const float* W_out    = (const float*)d_in[18];
  const float* b_out    = (const float*)d_in[19];
  const float* skip     = (const float*)d_in[20];

  constexpr int NQ = 8192, NA = 32768, NE = 500000;
  constexpr long long NT = (long long)NQ + NA + NE;
  const int Nt[3] = {NQ, NA, NE};
  const long long off[3] = {0, NQ, (long long)NQ + NA};
  const int Er[4] = {131072, 800000, 262144, 32768};
  const int st_[4] = {2, 2, 2, 0};
  const int dt_[4] = {0, 2, 1, 1};

  // workspace layout (floats): [h NT*64][kqv NT*192][agg NT*64][s NT*2][stats 768][m NT*2 u32][sc E*2]
  float*    f_h   = (float*)d_ws;
  float*    f_kqv = f_h + NT * 64;
  float*    f_agg = f_kqv + NT * 192;
  float*    f_s   = f_agg + NT * 64;
  float*    f_st  = f_s + NT * 2;       // sum[192] sumsq[192] mu[192] rs[192]
  unsigned* f_m   = (unsigned*)(f_st + 768);
  float*    f_sc  = (float*)(f_m + NT * 2);
  long long scoff[4];
  scoff[0] = 0;
  for (int r = 1; r < 4; ++r) scoff[r] = scoff[r - 1] + Er[r - 1];

  float* outp = (float*)d_out;
  float* pos_out = outp;
  float* neg_out = outp + 8192;
  float* z[3];
  z[0] = outp + 32768;                  // zq
  z[1] = z[0] + (long long)NQ * 64;     // za
  z[2] = z[1] + (long long)NA * 64;     // ze

  // init: zeros for agg+s+stats (contiguous), -inf-encoding for m
  {
    size_t nz = (size_t)(NT * 64 + NT * 2 + 768);
    k_fill_u32<<<(unsigned)((nz + 255) / 256), 256, 0, stream>>>((unsigned*)f_agg, 0u, nz);
    size_t nm = (size_t)(NT * 2);
    k_fill_u32<<<(unsigned)((nm + 255) / 256), 256, 0, stream>>>(f_m, 0x007FFFFFu, nm);
  }

  const float* xs[3] = {xq, xa, xe};
  for (int t = 0; t < 3; ++t)
    k_in_proj<<<Nt[t] / 16, 128, 0, stream>>>(
        xs[t], W_in + (long long)t * 128 * 64, b_in + t * 64, f_h + off[t] * 64, Nt[t]);

  for (int t = 0; t < 3; ++t)
    k_bn_reduce<<<256, 256, 0, stream>>>(f_h + off[t] * 64, Nt[t], f_st + t * 64, f_st + 192 + t * 64);
  k_bn_finalize<<<1, 192, 0, stream>>>(f_st, f_st + 192, f_st + 384, f_st + 576, NQ, NA, NE);
  for (int t = 0; t < 3; ++t) {
    long long n = (long long)Nt[t] * 64;
    k_bn_apply<<<(unsigned)((n + 255) / 256), 256, 0, stream>>>(
        f_h + off[t] * 64, n, bn_gamma + t * 64, bn_beta + t * 64, f_st + 384 + t * 64, f_st + 576 + t * 64);
  }

  for (int t = 0; t < 3; ++t)
    k_kqv<<<dim3(Nt[t] / 16, 3), 128, 0, stream>>>(
        f_h + off[t] * 64, W_kqv + (long long)t * 64 * 192, b_kqv + t * 192, f_kqv + off[t] * 192, Nt[t]);

  for (int r = 0; r < 4; ++r) {
    int E = Er[r];
    k_edge_pass1<<<(E + 3) / 4, 256, 0, stream>>>(
        eptr[r], eptr[r] + E, E,
        f_kqv + off[st_[r]] * 192, f_kqv + off[dt_[r]] * 192,
        A_k + (long long)r * 2048, p_rel + r * 2,
        f_sc + scoff[r] * 2, f_m + off[dt_[r]] * 2);
  }
  for (int r = 0; r < 4; ++r) {
    int E = Er[r];
    k_edge_pass2<<<(E + 3) / 4, 256, 0, stream>>>(
        eptr[r], eptr[r] + E, E,
        f_kqv + off[st_[r]] * 192, A_v + (long long)r * 2048,
        f_sc + scoff[r] * 2, f_m + off[dt_[r]] * 2,
        f_s + off[dt_[r]] * 2, f_agg + off[dt_[r]] * 64);
  }

  for (int t = 0; t < 3; ++t)
    k_out<<<Nt[t] / 16, 128, 0, stream>>>(
        f_agg + off[t] * 64, f_s + off[t] * 2, f_h + off[t] * 64,
        W_out + (long long)t * 4096, b_out + t * 64, skip + t, z[t], Nt[t]);

  k_pred<<<(8192 + 255) / 256, 256, 0, stream>>>(pos_idx, pos_idx + 8192, 8192, z[0], z[1], pos_out);
  k_pred<<<(24576 + 255) / 256, 256, 0, stream>>>(neg_idx, neg_idx + 24576, 24576, z[0], z[1], neg_out);
}